// KDNet_81673098101054
// MI455X (gfx1250) — compile-verified
//
#include <hip/hip_runtime.h>
#include <math.h>

// ---------------------------------------------------------------------------
// KD-Net forward for MI455X (gfx1250, wave32).
// Memory-bound (~30 GFLOP vs ~1.2 GB Y traffic @ 23.3 TB/s) -> exact fp32
// WMMA path (V_WMMA_F32_16X16X4_F32). GEMM staging uses CDNA5 async
// global->LDS loads (ASYNCcnt) with double-buffered LDS.
// ---------------------------------------------------------------------------

typedef float v2f __attribute__((ext_vector_type(2)));
typedef float v8f __attribute__((ext_vector_type(8)));

#define EPSV 1e-5f
#define BM 32
#define BN 64

// low 32 bits of a generic pointer to __shared__ == wave-relative LDS offset
__device__ __forceinline__ unsigned lds_off(const void* p) {
  return (unsigned)(unsigned long long)p;
}
__device__ __forceinline__ void async_load_b32(unsigned ldsaddr, const void* gaddr) {
  asm volatile("global_load_async_to_lds_b32 %0, %1, off"
               :: "v"(ldsaddr), "v"(gaddr) : "memory");
}
__device__ __forceinline__ void wait_async0() {
  asm volatile("s_wait_asynccnt 0x0" ::: "memory");
}

// ---- fc1: h[b,f,n] = relu(sum_c x[b,c,n] * w[f,c] + b[f]), f in [0,8) ------
__global__ void fc1_kernel(const float* __restrict__ x, const float* __restrict__ w,
                           const float* __restrict__ bias, float* __restrict__ h,
                           int total, int N) {
  int idx = blockIdx.x * blockDim.x + threadIdx.x;
  if (idx >= total) return;
  int b = idx / N, n = idx % N;
  float x0 = x[(b * 3 + 0) * N + n];
  float x1 = x[(b * 3 + 1) * N + n];
  float x2 = x[(b * 3 + 2) * N + n];
#pragma unroll
  for (int f = 0; f < 8; ++f) {
    float v = fmaf(w[f * 3 + 0], x0, fmaf(w[f * 3 + 1], x1, w[f * 3 + 2] * x2)) + bias[f];
    h[(b * 8 + f) * N + n] = fmaxf(v, 0.f);
  }
}

// ---- kdconv GEMM: Y[M x Ncols] = W[M x K] * Xc + bias ---------------------
// Xc[k, b*nh+j] = hprev[b, k mod fin, 2j + k div fin]   (K = 2*fin)
// KCT in {16,32}; requires K % KCT == 0 (guaranteed by dispatch) -> no guards.
template <int KCT>
__global__ __launch_bounds__(256)
void kdconv_gemm_wmma(const float* __restrict__ Wm, const float* __restrict__ bias,
                      const float* __restrict__ hprev, float* __restrict__ Y,
                      int M, int K, int Ncols, int fin, int finShift,
                      int nh, int nhShift) {
  constexpr int AELEM = (BM * KCT) / 256;     // per-thread A elems per chunk
  constexpr int BELEM = (KCT * BN) / 256;     // per-thread B elems per chunk
  constexpr int ASTRT = KCT + 2;              // even stride: 8B-aligned f2 frags
  constexpr unsigned ABUF = BM * ASTRT * 4;   // bytes per A buffer
  constexpr unsigned BBUF = KCT * BN * 4;     // bytes per B buffer

  __shared__ float As[2][BM * ASTRT];
  __shared__ float Bs[2][KCT * BN];

  const int tid  = threadIdx.x;
  const int lane = tid & 31;
  const int wave = tid >> 5;                 // 8 waves: 2(M) x 4(N)
  const int wm   = wave >> 2;
  const int wn   = wave & 3;
  const int mBase0 = blockIdx.y * BM;
  const int nBase0 = blockIdx.x * BN;
  const int n2 = nh << 1;

  const int laneN  = lane & 15;
  const int hiHalf = lane >> 4;

  // --- per-thread staging indices / LDS offsets (buffer 0) -----------------
  int aml[AELEM], akl[AELEM];
  unsigned aOff[AELEM];
#pragma unroll
  for (int i = 0; i < AELEM; ++i) {
    int e = tid + i * 256;
    aml[i] = e / KCT;
    akl[i] = e % KCT;
    aOff[i] = lds_off(&As[0][aml[i] * ASTRT + akl[i]]);
  }
  int bkl[BELEM];
  unsigned bOff[BELEM];
  size_t bbase[BELEM];                        // gather base: b*fin*n2 + 2j
#pragma unroll
  for (int i = 0; i < BELEM; ++i) {
    int e  = tid + i * 256;
    bkl[i] = e / BN;
    int cl = e % BN;
    bOff[i] = lds_off(&Bs[0][bkl[i] * BN + cl]);
    int col = nBase0 + cl;
    int b   = col >> nhShift;
    int j   = col & (nh - 1);
    bbase[i] = (size_t)(b * fin) * n2 + (j << 1);
  }

  const int nchunk = K / KCT;

  // --- prologue: async-stage chunk 0 into buffer 0 -------------------------
#pragma unroll
  for (int i = 0; i < AELEM; ++i)
    async_load_b32(aOff[i], Wm + (size_t)(mBase0 + aml[i]) * K + akl[i]);
#pragma unroll
  for (int i = 0; i < BELEM; ++i) {
    int kg = bkl[i];
    async_load_b32(bOff[i],
        hprev + bbase[i] + (size_t)(kg & (fin - 1)) * n2 + (kg >> finShift));
  }
  wait_async0();
  __syncthreads();

  v8f acc = {};

  for (int ck = 0; ck < nchunk; ++ck) {
    const int buf = ck & 1;

    // async-stage next chunk into the other buffer (lands during WMMA block)
    if (ck + 1 < nchunk) {
      const int k0 = (ck + 1) * KCT;
      const unsigned aSel = (1 - buf) * ABUF;
      const unsigned bSel = (1 - buf) * BBUF;
#pragma unroll
      for (int i = 0; i < AELEM; ++i)
        async_load_b32(aOff[i] + aSel, Wm + (size_t)(mBase0 + aml[i]) * K + k0 + akl[i]);
#pragma unroll
      for (int i = 0; i < BELEM; ++i) {
        int kg = k0 + bkl[i];
        async_load_b32(bOff[i] + bSel,
            hprev + bbase[i] + (size_t)(kg & (fin - 1)) * n2 + (kg >> finShift));
      }
    }

    // compute on current buffer: KCT/4 WMMAs (EXEC all-ones, no divergence)
#pragma unroll
    for (int kk = 0; kk < KCT; kk += 4) {
      int ka = kk + (hiHalf << 1);
      const float* ap = &As[buf][(wm * 16 + laneN) * ASTRT + ka];
      v2f a; a.x = ap[0]; a.y = ap[1];
      const float* bp = &Bs[buf][ka * BN + wn * 16 + laneN];
      v2f bf; bf.x = bp[0]; bf.y = bp[BN];
      acc = __builtin_amdgcn_wmma_f32_16x16x4_f32(
          false, a, false, bf, (short)0, acc, false, false);
    }

    // our async writes must be complete before anyone reads the next buffer
    wait_async0();
    __syncthreads();
  }

  // epilogue: bias + store (C/D layout: VGPR r -> M = r + 8*hiHalf, N = laneN)
  const int ch0 = mBase0 + wm * 16 + hiHalf * 8;
  const int col = nBase0 + wn * 16 + laneN;
#pragma unroll
  for (int r = 0; r < 8; ++r) {
    int ch = ch0 + r;
    Y[(size_t)ch * Ncols + col] = acc[r] + bias[ch];
  }
}

// ---- per-channel mean / rsqrt(var+eps) over all Ncols ----------------------
__global__ __launch_bounds__(256)
void bn_stats_kernel(const float* __restrict__ Y, float* __restrict__ stats, int Ncols) {
  const int ch = blockIdx.x;
  const float4* row4 = (const float4*)(Y + (size_t)ch * Ncols);
  const int n4 = Ncols >> 2;               // Ncols is always a multiple of 4
  float s = 0.f, sq = 0.f;
  for (int i = threadIdx.x; i < n4; i += 256) {
    float4 v = row4[i];
    s += v.x + v.y + v.z + v.w;
    sq = fmaf(v.x, v.x, fmaf(v.y, v.y, fmaf(v.z, v.z, fmaf(v.w, v.w, sq))));
  }
  __shared__ float ss[256], qq[256];
  ss[threadIdx.x] = s;
  qq[threadIdx.x] = sq;
  __syncthreads();
  for (int off = 128; off > 0; off >>= 1) {
    if (threadIdx.x < off) {
      ss[threadIdx.x] += ss[threadIdx.x + off];
      qq[threadIdx.x] += qq[threadIdx.x + off];
    }
    __syncthreads();
  }
  if (threadIdx.x == 0) {
    float inv  = 1.f / (float)Ncols;
    float mean = ss[0] * inv;
    float var  = qq[0] * inv - mean * mean;
    stats[2 * ch]     = mean;
    stats[2 * ch + 1] = rsqrtf(var + EPSV);
  }
}

// ---- normalize + affine + relu + kd-select --------------------------------
__global__ void bn_select_kernel(const float* __restrict__ Y, const float* __restrict__ stats,
                                 const float* __restrict__ g, const float* __restrict__ beta,
                                 const int* __restrict__ sel, float* __restrict__ out,
                                 int fout, int nh, int Ncols, int total) {
  int idx = blockIdx.x * blockDim.x + threadIdx.x;
  if (idx >= total) return;
  int j = idx % nh;
  int t = idx / nh;
  int f = t % fout;
  int b = t / fout;
  int s   = sel[b * nh + j];               // 0..2
  int ch  = s * fout + f;
  int col = b * nh + j;
  float v = Y[(size_t)ch * Ncols + col];
  v = (v - stats[2 * ch]) * stats[2 * ch + 1] * g[ch] + beta[ch];
  out[idx] = fmaxf(v, 0.f);
}

// ---- head: scores = feat @ fc8_w^T + b, then log_softmax ------------------
__global__ __launch_bounds__(256)
void head_kernel(const float* __restrict__ feat, const float* __restrict__ w,
                 const float* __restrict__ bias, float* __restrict__ out) {
  __shared__ float f[1024];
  __shared__ float sc[64];
  const int b    = blockIdx.x;
  const int lane = threadIdx.x & 31;
  const int wave = threadIdx.x >> 5;       // 8 waves
  for (int i = threadIdx.x; i < 1024; i += 256) f[i] = feat[(size_t)b * 1024 + i];
  __syncthreads();
  for (int cls = wave; cls < 40; cls += 8) {
    const float* wr = w + (size_t)cls * 1024;
    float s = 0.f;
    for (int k = lane; k < 1024; k += 32) s = fmaf(wr[k], f[k], s);
#pragma unroll
    for (int off = 16; off > 0; off >>= 1) s += __shfl_xor(s, off, 32);
    if (lane == 0) sc[cls] = s + bias[cls];
  }
  __syncthreads();
  if (threadIdx.x == 0) {
    float m = -INFINITY;
    for (int i = 0; i < 40; ++i) m = fmaxf(m, sc[i]);
    float sum = 0.f;
    for (int i = 0; i < 40; ++i) sum += __expf(sc[i] - m);
    sc[40] = m;
    sc[41] = __logf(sum);
  }
  __syncthreads();
  if (threadIdx.x < 40)
    out[(size_t)b * 40 + threadIdx.x] = sc[threadIdx.x] - sc[40] - sc[41];
}

// ---------------------------------------------------------------------------
static const int FEATS[11] = {8, 32, 64, 64, 128, 128, 256, 256, 512, 512, 1024};

extern "C" void kernel_launch(void* const* d_in, const int* in_sizes, int n_in,
                              void* d_out, int out_size, void* d_ws, size_t ws_size,
                              hipStream_t stream) {
  (void)in_sizes; (void)n_in; (void)out_size; (void)ws_size;
  const int B = 512, N = 1024;

  const float* x = (const float*)d_in[0];
  const int* sel[10];
  for (int i = 0; i < 10; ++i) sel[i] = (const int*)d_in[1 + i];
  const float* fc1_w = (const float*)d_in[11];
  const float* fc1_b = (const float*)d_in[12];
  const float *Wp[10], *bp[10], *gp[10], *betap[10];
  for (int i = 0; i < 10; ++i) {
    Wp[i]    = (const float*)d_in[13 + 4 * i];
    bp[i]    = (const float*)d_in[14 + 4 * i];
    gp[i]    = (const float*)d_in[15 + 4 * i];
    betap[i] = (const float*)d_in[16 + 4 * i];
  }
  const float* fc8_w = (const float*)d_in[53];
  const float* fc8_b = (const float*)d_in[54];

  // workspace layout
  char* ws = (char*)d_ws;
  const size_t Ybytes = (size_t)96 * 262144 * 4;    // 100.66 MB (level-0 Y, max)
  const size_t Hbytes = (size_t)512 * 32 * 512 * 4; // 33.55 MB  (max h)
  float* Y     = (float*)ws;
  float* hA    = (float*)(ws + Ybytes);
  float* hB    = (float*)(ws + Ybytes + Hbytes);
  float* stats = (float*)(ws + Ybytes + 2 * Hbytes); // 2*3072 floats

  // fc1
  {
    int total = B * N;
    fc1_kernel<<<(total + 255) / 256, 256, 0, stream>>>(x, fc1_w, fc1_b, hA, total, N);
  }

  float* hcur = hA;
  float* hnext = hB;
  for (int i = 0; i < 10; ++i) {
    const int fin = FEATS[i], fout = FEATS[i + 1];
    const int M = 3 * fout, K = 2 * fin;
    const int nh = (N >> i) >> 1;
    const int Ncols = B * nh;
    const int finShift = __builtin_ctz(fin);
    const int nhShift  = __builtin_ctz(nh);

    dim3 grid(Ncols / BN, M / BM);
    if ((K & 31) == 0) {
      kdconv_gemm_wmma<32><<<grid, 256, 0, stream>>>(Wp[i], bp[i], hcur, Y,
                                                     M, K, Ncols, fin, finShift, nh, nhShift);
    } else {  // level 0: K = 16
      kdconv_gemm_wmma<16><<<grid, 256, 0, stream>>>(Wp[i], bp[i], hcur, Y,
                                                     M, K, Ncols, fin, finShift, nh, nhShift);
    }
    bn_stats_kernel<<<M, 256, 0, stream>>>(Y, stats, Ncols);
    const int total = B * fout * nh;
    bn_select_kernel<<<(total + 255) / 256, 256, 0, stream>>>(
        Y, stats, gp[i], betap[i], sel[i], hnext, fout, nh, Ncols, total);
    float* t = hcur; hcur = hnext; hnext = t;
  }

  // after level 9: hcur is (B, 1024, 1) == (B, 1024) contiguous
  head_kernel<<<B, 256, 0, stream>>>(hcur, fc8_w, fc8_b, (float*)d_out);
}